// ARoIAlignRotated_70196945486205
// MI455X (gfx1250) — compile-verified
//
#include <hip/hip_runtime.h>
#include <math.h>

// ---------------------------------------------------------------------------
// RoI Align Rotated for MI455X (gfx1250, wave32).
//
// Strategy:
//   * One wave per (roi, bin). 49 bins x 1000 rois = 49,000 waves.
//   * Per wave: compute the 4 rotated sample points, their bilinear weights
//     (x 1/4 sample-mean, x valid mask) and clamped corner indices ONCE.
//   * Channel loop (256 ch in 16 groups of 16): each lane gathers 2 corner
//     values per sample (half-wave 0 = y0 row, half-wave 1 = y1 row), and the
//     16-term weighted reduction  out[c] = sum_k w_k * F[k,c]  is done on the
//     matrix pipe with 4 accumulating V_WMMA_F32_16X16X4_F32 ops:
//        A(16x4)  = sample weights, replicated across M (wave-uniform)
//        B(4x16)  = gathered corner features for 16 channels
//     A-layout (ISA 7.12.2): lanes 0-15 carry K=0,1 / lanes 16-31 carry K=2,3,
//     which matches the half-wave split (corners y0x0,y0x1 vs y1x0,y1x1).
//     All D rows are identical; lanes 0-15 store accumulator VGPR0 = 16 ch.
//   * Gather-bound kernel: features (134 MB) fit in the 192 MB L2, so the
//     ~800 MB of 4B gathers are mostly L2 hits; WMMA keeps VALU slots free
//     for address math, 8 waves/block x 6125 blocks hide latency.
// ---------------------------------------------------------------------------

typedef __attribute__((ext_vector_type(2))) float v2f;
typedef __attribute__((ext_vector_type(8))) float v8f;

#define CCH   256      // channels
#define HH    256      // feature height
#define WW    256      // feature width
#define OUT_H 7
#define OUT_W 7
#define NBINS (OUT_H * OUT_W)     // 49
#define WAVES_PER_BLOCK 8
#define SPATIAL_SCALE 0.25f

__global__ __launch_bounds__(256) void ARoIAlignRotated_wmma_kernel(
    const float* __restrict__ feat,   // (B, C, H, W)
    const float* __restrict__ rois,   // (N, 6): b, cx, cy, w, h, theta
    float* __restrict__ out,          // (N, C, OUT_H, OUT_W)
    int N)
{
    const int lane = threadIdx.x & 31;
    const int wave = threadIdx.x >> 5;
    const int bin  = blockIdx.x * WAVES_PER_BLOCK + wave;   // wave-uniform
    if (bin >= N * NBINS) return;                           // wave-uniform exit

    const int n  = bin / NBINS;
    const int pb = bin - n * NBINS;
    const int ph = pb / OUT_W;
    const int pw = pb - ph * OUT_W;

    // ---- roi parameters (wave-uniform; broadcast loads) -------------------
    const float* r = rois + (size_t)n * 6;
    const int   bidx = (int)r[0];
    const float cx   = r[1] * SPATIAL_SCALE;
    const float cy   = r[2] * SPATIAL_SCALE;
    const float rw   = fmaxf(r[3] * SPATIAL_SCALE, 0.0f);
    const float rh   = fmaxf(r[4] * SPATIAL_SCALE, 0.0f);
    const float th   = r[5];
    const float bin_h = rh * (1.0f / OUT_H);
    const float bin_w = rw * (1.0f / OUT_W);
    const float cs = cosf(th);
    const float sn = sinf(th);

    // half-wave 0 handles corners (y0,x0),(y0,x1); half 1 handles (y1,*)
    const int half = lane >> 4;

    // ---- per-sample corner indices + A-matrix weights ---------------------
    int   idx0[4], idx1[4];
    float wa0[4],  wa1[4];
#pragma unroll
    for (int s = 0; s < 4; ++s) {
        const int iy = s >> 1, ix = s & 1;
        const float yy = -rh * 0.5f + bin_h * ((float)ph + ((float)iy + 0.5f) * 0.5f);
        const float xx = -rw * 0.5f + bin_w * ((float)pw + ((float)ix + 0.5f) * 0.5f);
        const float y  = yy * cs - xx * sn + cy;
        const float x  = yy * sn + xx * cs + cx;
        const bool valid = (y > -1.0f) && (y < (float)HH) &&
                           (x > -1.0f) && (x < (float)WW);
        const float yc = fminf(fmaxf(y, 0.0f), (float)(HH - 1));
        const float xc = fminf(fmaxf(x, 0.0f), (float)(WW - 1));
        int y0 = (int)yc; if (y0 > HH - 1) y0 = HH - 1;     // yc >= 0 -> trunc == floor
        int x0 = (int)xc; if (x0 > WW - 1) x0 = WW - 1;
        const int y1 = (y0 + 1 < HH) ? y0 + 1 : HH - 1;
        const int x1 = (x0 + 1 < WW) ? x0 + 1 : WW - 1;
        const float ly = yc - (float)y0, lx = xc - (float)x0;
        const float hy = 1.0f - ly,      hx = 1.0f - lx;
        const float vm = valid ? 0.25f : 0.0f;              // fold 1/4 sample mean

        const int row = half ? y1 : y0;
        idx0[s] = row * WW + x0;
        idx1[s] = row * WW + x1;
        const float wy = half ? ly : hy;                    // w00/w01 vs w10/w11
        wa0[s] = wy * hx * vm;
        wa1[s] = wy * lx * vm;
    }

    // ---- channel loop: gather B, 4 accumulating WMMAs, store 16 channels --
    const int    c0    = lane & 15;                          // channel within group
    const float* base  = feat + ((size_t)bidx * CCH + c0) * (size_t)(HH * WW);
    float*       obase = out + ((size_t)n * CCH + c0) * NBINS + pb;

#pragma unroll 1
    for (int g = 0; g < CCH / 16; ++g) {
        const float* p = base + (size_t)g * 16 * (HH * WW);
        v8f acc = {};
#pragma unroll
        for (int s = 0; s < 4; ++s) {
            v2f a, b;
            a.x = wa0[s];          // A[M, K=2*half+0]  (rows replicated)
            a.y = wa1[s];          // A[M, K=2*half+1]
            b.x = p[idx0[s]];      // B[K=2*half+0, N=c0]
            b.y = p[idx1[s]];      // B[K=2*half+1, N=c0]
            // 8 args: (neg_a, A, neg_b, B, c_mod, C, reuse_a, reuse_b)
            acc = __builtin_amdgcn_wmma_f32_16x16x4_f32(
                false, a, false, b, (short)0, acc, false, false);
        }
        // All D rows identical; lanes 0-15 VGPR0 hold out[c_base + 0..15].
        if (lane < 16) obase[(size_t)g * 16 * NBINS] = acc[0];
    }
}

extern "C" void kernel_launch(void* const* d_in, const int* in_sizes, int n_in,
                              void* d_out, int out_size, void* d_ws, size_t ws_size,
                              hipStream_t stream) {
    const float* feat = (const float*)d_in[0];   // (2,256,256,256) f32
    const float* rois = (const float*)d_in[1];   // (N,6) f32
    float* out = (float*)d_out;                  // (N,256,7,7) f32

    const int N = in_sizes[1] / 6;
    const int totalWaves = N * NBINS;
    const int blocks = (totalWaves + WAVES_PER_BLOCK - 1) / WAVES_PER_BLOCK;

    ARoIAlignRotated_wmma_kernel<<<blocks, WAVES_PER_BLOCK * 32, 0, stream>>>(
        feat, rois, out, N);
}